// TempHGT_24489903522224
// MI455X (gfx1250) — compile-verified
//
#include <hip/hip_runtime.h>
#include <math.h>

#define NHID 256
#define DKv  32
#define HEADS 8

typedef float v2f __attribute__((ext_vector_type(2)));
typedef float v8f __attribute__((ext_vector_type(8)));

// order-preserving float<->uint transform for atomicMax on floats
static __device__ __forceinline__ unsigned int f2ord(float f) {
  unsigned int u = __float_as_uint(f);
  return (u & 0x80000000u) ? ~u : (u | 0x80000000u);
}
static __device__ __forceinline__ float ord2f(unsigned int u) {
  return (u & 0x80000000u) ? __uint_as_float(u & 0x7FFFFFFFu) : __uint_as_float(~u);
}
#define ORD_NEG_INF 0x007FFFFFu  // f2ord(-inf)

// ---------------- utility fills ----------------
__global__ void fill_f32_k(float* __restrict__ p, float v, long n) {
  long i = (long)blockIdx.x * blockDim.x + threadIdx.x;
  if (i < n) p[i] = v;
}
__global__ void fill_u32_k(unsigned int* __restrict__ p, unsigned int v, long n) {
  long i = (long)blockIdx.x * blockDim.x + threadIdx.x;
  if (i < n) p[i] = v;
}

// th_tab[t][j] = time_tab[t] @ time_w + time_b   (7 x 32)
__global__ void time_enc_k(const float* __restrict__ tab, const float* __restrict__ w,
                           const float* __restrict__ b, float* __restrict__ out) {
  int t = threadIdx.x;
  if (t >= 7 * 32) return;
  int ti = t >> 5, j = t & 31;
  float s = b[j];
  for (int i = 0; i < 32; ++i) s += tab[ti * 32 + i] * w[i * 32 + j];
  out[t] = s;
}

// dst[r][c] = src[idx ? idx[r] : 0][c]
__global__ void gather_rows_k(const float* __restrict__ src, const int* __restrict__ idx,
                              float* __restrict__ dst, int n) {
  long t = (long)blockIdx.x * blockDim.x + threadIdx.x;
  if (t >= (long)n * NHID) return;
  int r = (int)(t >> 8), c = (int)(t & 255);
  int sr = idx ? idx[r] : 0;
  dst[t] = src[(size_t)sr * NHID + c];
}

// ---------------- WMMA f32 GEMM: C[M,256] = scaleA*A'[M,256] @ B[256,256] + bias ----------------
// A' = idx ? A[idx[m]] : A[m].  grid = (4, ceil(M/128)), block = 256 (8 waves, 16x64 tile each).
// The 256x64 B strip is staged once per block into LDS in [kgroup][col][4] layout so each
// lane's (k,k+1) B pair is one conflict-free ds_load_b64 (all 64 banks hit exactly once).
__global__ __launch_bounds__(256) void gemm256_wmma_k(
    const float* __restrict__ A, const int* __restrict__ idx,
    const float* __restrict__ B, const float* __restrict__ bias,
    float* __restrict__ C, int M, float scaleA) {
  __shared__ float Bs[64 * 64 * 4];  // 64 kgroups x 64 cols x 4 k-in-group = 64KB
  const int tid  = threadIdx.x;
  const int col0 = blockIdx.x * 64;

  // cooperative stage: per (g,c): Bs[(g*64+c)*4 + j] = B[4g+j][col0+c]
  {
    const int c = tid & 63;
    for (int g = tid >> 6; g < 64; g += 4) {
      const float* Bp = B + (size_t)(4 * g) * NHID + col0 + c;
      float4 v;
      v.x = Bp[0]; v.y = Bp[NHID]; v.z = Bp[2 * NHID]; v.w = Bp[3 * NHID];
      *(float4*)&Bs[(g * 64 + c) * 4] = v;
    }
  }
  __syncthreads();

  const int wave = tid >> 5;
  const int lane = tid & 31;
  const int lh = lane & 15;
  const bool hi = lane >= 16;
  const int row0 = (blockIdx.y * 8 + wave) * 16;
  if (row0 >= M) return;  // uniform per wave (after the only barrier) -> EXEC all-1s for WMMA
  int arow = row0 + lh; if (arow >= M) arow = M - 1;
  const int gr = idx ? idx[arow] : arow;
  const float* Ar = A + (size_t)gr * NHID;
  const int koff = hi ? 2 : 0;

  v8f acc[4] = {};
  for (int g = 0; g < 64; ++g) {
    v2f a = *(const v2f*)(Ar + 4 * g + koff);  // A[row][4g+koff .. +1]
#pragma unroll
    for (int t = 0; t < 4; ++t) {
      v2f b = *(const v2f*)&Bs[(g * 64 + t * 16 + lh) * 4 + koff];
      acc[t] = __builtin_amdgcn_wmma_f32_16x16x4_f32(
          false, a, false, b, (short)0, acc[t], false, false);
    }
  }
#pragma unroll
  for (int t = 0; t < 4; ++t) {
    const int col = col0 + t * 16 + lh;
    const float bb = bias ? bias[col] : 0.0f;
#pragma unroll
    for (int r = 0; r < 8; ++r) {
      const int m = row0 + r + (hi ? 8 : 0);
      if (m < M) C[(size_t)m * NHID + col] = acc[t][r] * scaleA + bb;
    }
  }
}

// ---------------- per-head 32x32 WMMA: Y[n, h*32+j] = sum_i X[n, h*32+i] * W(h,i,j) ----------------
// transW=1 -> W(h,i,j) = W[h][j][i] (q @ attR^T); transpose folded into LDS staging.
// grid = (ceil(n/128), 8), block = 256.
__global__ __launch_bounds__(256) void headgemm_wmma_k(
    const float* __restrict__ X, const float* __restrict__ W,
    float* __restrict__ Y, int n, int transW) {
  __shared__ float Ws[8 * 32 * 4];  // 8 kgroups x 32 cols x 4 = 4KB
  const int tid = threadIdx.x;
  const int h = blockIdx.y;
  const float* Wh = W + (size_t)h * DKv * DKv;
  {
    const int g = tid >> 5, c = tid & 31;  // 8 x 32 = 256 threads, one (g,c) each
    float4 v;
    if (transW) {  // contiguous read of Wh[c][4g..4g+3]
      const float* p = Wh + c * DKv + 4 * g;
      v.x = p[0]; v.y = p[1]; v.z = p[2]; v.w = p[3];
    } else {       // strided read of Wh[4g+j][c]
      const float* p = Wh + (size_t)(4 * g) * DKv + c;
      v.x = p[0]; v.y = p[DKv]; v.z = p[2 * DKv]; v.w = p[3 * DKv];
    }
    *(float4*)&Ws[(g * 32 + c) * 4] = v;
  }
  __syncthreads();

  const int wave = tid >> 5;
  const int lane = tid & 31;
  const int lh = lane & 15;
  const bool hi = lane >= 16;
  const int row0 = (blockIdx.x * 8 + wave) * 16;
  if (row0 >= n) return;
  int arow = row0 + lh; if (arow >= n) arow = n - 1;
  const float* Xr = X + (size_t)arow * NHID + h * DKv;
  const int koff = hi ? 2 : 0;

  v8f acc[2] = {};
#pragma unroll
  for (int g = 0; g < 8; ++g) {
    v2f a = *(const v2f*)(Xr + 4 * g + koff);
#pragma unroll
    for (int t = 0; t < 2; ++t) {
      v2f b = *(const v2f*)&Ws[(g * 32 + t * 16 + lh) * 4 + koff];
      acc[t] = __builtin_amdgcn_wmma_f32_16x16x4_f32(
          false, a, false, b, (short)0, acc[t], false, false);
    }
  }
#pragma unroll
  for (int t = 0; t < 2; ++t) {
    const int col = t * 16 + lh;
#pragma unroll
    for (int r = 0; r < 8; ++r) {
      const int m = row0 + r + (hi ? 8 : 0);
      if (m < n) Y[(size_t)m * NHID + h * DKv + col] = acc[t][r];
    }
  }
}

// ---------------- edge passes ----------------
// a[e,h] = dot(k[src,h,:], qa[dst,h,:]) * pri[h] / sqrt(32);  segment max via atomics
__global__ void edge_score_k(const float* __restrict__ ksrc, const float* __restrict__ qa,
                             const int* __restrict__ src, const int* __restrict__ dst,
                             const float* __restrict__ pri, float* __restrict__ aedge,
                             unsigned int* __restrict__ mbuf, int E) {
  long t = (long)blockIdx.x * blockDim.x + threadIdx.x;
  if (t >= (long)E * HEADS) return;
  int e = (int)(t >> 3), h = (int)(t & 7);
  const float* kr = ksrc + (size_t)src[e] * NHID + h * DKv;
  const float* qr = qa + (size_t)dst[e] * NHID + h * DKv;
  float a = 0.0f;
#pragma unroll 8
  for (int i = 0; i < DKv; ++i) a += kr[i] * qr[i];
  a *= pri[h] * 0.17677669529663687f;  // 1/sqrt(32)
  aedge[t] = a;
  atomicMax(&mbuf[(size_t)dst[e] * HEADS + h], f2ord(a));
}

// ex = exp(a - m[dst]);  den[dst] += ex   (empty-segment: m<-1e30 -> 0, matches reference)
__global__ void edge_exp_k(const int* __restrict__ dst, const unsigned int* __restrict__ mbuf,
                           float* __restrict__ aedge, float* __restrict__ den, int E) {
  long t = (long)blockIdx.x * blockDim.x + threadIdx.x;
  if (t >= (long)E * HEADS) return;
  int e = (int)(t >> 3), h = (int)(t & 7);
  float m = ord2f(mbuf[(size_t)dst[e] * HEADS + h]);
  if (!(m > -1e30f)) m = 0.0f;
  float ex = __expf(aedge[t] - m);
  aedge[t] = ex;
  atomicAdd(&den[(size_t)dst[e] * HEADS + h], ex);
}

// agg[dst,c] += (ex/den) * (vm[src,c] + th[time[e], c%32])
__global__ void edge_agg_k(const int* __restrict__ src, const int* __restrict__ dst,
                           const int* __restrict__ time, const float* __restrict__ tht,
                           const float* __restrict__ vm, const float* __restrict__ aedge,
                           const float* __restrict__ den, float* __restrict__ agg, int E) {
  long t = (long)blockIdx.x * blockDim.x + threadIdx.x;
  if (t >= (long)E * NHID) return;
  int e = (int)(t >> 8), c = (int)(t & 255), h = c >> 5;
  float w = aedge[(size_t)e * HEADS + h] /
            fmaxf(den[(size_t)dst[e] * HEADS + h], 1e-9f);
  float val = vm[(size_t)src[e] * NHID + c];
  if (time) val += tht[time[e] * 32 + (c & 31)];
  atomicAdd(&agg[(size_t)dst[e] * NHID + c], w * val);
}

__global__ void relu_acc_k(const float* __restrict__ agg, float* __restrict__ tacc, long n) {
  long i = (long)blockIdx.x * blockDim.x + threadIdx.x;
  if (i < n) tacc[i] += fmaxf(agg[i], 0.0f);
}

// out = trans*alpha + h*(1-alpha); LayerNorm over 256; one block per row
__global__ __launch_bounds__(256) void skip_ln_k(
    const float* __restrict__ trans, const float* __restrict__ h,
    const float* __restrict__ skip, const float* __restrict__ g,
    const float* __restrict__ b, float* __restrict__ nh) {
  __shared__ float red[NHID];
  const int row = blockIdx.x, c = threadIdx.x;
  const float alpha = 1.0f / (1.0f + __expf(-skip[0]));
  const float o = trans[(size_t)row * NHID + c] * alpha +
                  h[(size_t)row * NHID + c] * (1.0f - alpha);
  red[c] = o; __syncthreads();
  for (int s = 128; s > 0; s >>= 1) { if (c < s) red[c] += red[c + s]; __syncthreads(); }
  const float mu = red[0] * (1.0f / NHID); __syncthreads();
  const float dv = o - mu;
  red[c] = dv * dv; __syncthreads();
  for (int s = 128; s > 0; s >>= 1) { if (c < s) red[c] += red[c + s]; __syncthreads(); }
  const float var = red[0] * (1.0f / NHID);
  nh[(size_t)row * NHID + c] = dv * rsqrtf(var + 1e-5f) * g[c] + b[c];
}

__global__ void pool_max_k(const float* __restrict__ x, const int* __restrict__ grp,
                           unsigned int* __restrict__ pool, int n, int off) {
  long t = (long)blockIdx.x * blockDim.x + threadIdx.x;
  if (t >= (long)n * NHID) return;
  int r = (int)(t >> 8), c = (int)(t & 255);
  atomicMax(&pool[(size_t)grp[r] * (3 * NHID) + off + c], f2ord(x[t]));
}

// logits = gi @ out_w + out_b;  out[0]=BCE loss mean, out[1..G]=sigmoid(logits)
__global__ void final_k(const unsigned int* __restrict__ pool, const float* __restrict__ ow,
                        const float* __restrict__ ob, const float* __restrict__ y,
                        float* __restrict__ out, int Gn) {
  __shared__ float z[64];
  int tid = threadIdx.x;
  if (tid < Gn) {
    float s = ob[0];
    for (int c = 0; c < 3 * NHID; ++c) s += ord2f(pool[(size_t)tid * (3 * NHID) + c]) * ow[c];
    z[tid] = s;
    out[1 + tid] = 1.0f / (1.0f + __expf(-s));
  }
  __syncthreads();
  if (tid == 0) {
    float L = 0.0f;
    for (int gg = 0; gg < Gn; ++gg) {
      float zz = z[gg];
      L += fmaxf(zz, 0.0f) - zz * y[gg] + log1pf(__expf(-fabsf(zz)));
    }
    out[0] = L / (float)Gn;
  }
}

// ---------------- host orchestration ----------------
static inline int cdiv(long a, long b) { return (int)((a + b - 1) / b); }

extern "C" void kernel_launch(void* const* d_in, const int* in_sizes, int n_in,
                              void* d_out, int out_size, void* d_ws, size_t ws_size,
                              hipStream_t stream) {
  const int* word_id  = (const int*)d_in[0];
  const int* topic_id = (const int*)d_in[1];
  const int* g_word   = (const int*)d_in[2];
  const int* g_topic  = (const int*)d_in[3];
  const int* g_doc    = (const int*)d_in[4];
  const int* src_ww = (const int*)d_in[5];  const int* dst_ww = (const int*)d_in[6];  const int* time_ww = (const int*)d_in[7];
  const int* src_wd = (const int*)d_in[8];  const int* dst_wd = (const int*)d_in[9];  const int* time_wd = (const int*)d_in[10];
  const int* src_wt = (const int*)d_in[11]; const int* dst_wt = (const int*)d_in[12]; const int* time_wt = (const int*)d_in[13];
  const int* src_td = (const int*)d_in[14]; const int* dst_td = (const int*)d_in[15]; const int* time_td = (const int*)d_in[16];
  const int* src_tt = (const int*)d_in[17]; const int* dst_tt = (const int*)d_in[18];
  const float* y_data      = (const float*)d_in[19];
  const float* word_embeds = (const float*)d_in[20];
  const float* topic_embeds= (const float*)d_in[21];
  const float* doc_gen     = (const float*)d_in[22];
  const float* adapt_w = (const float*)d_in[23]; const float* adapt_b = (const float*)d_in[24];
  const float* time_tab= (const float*)d_in[25]; const float* time_w  = (const float*)d_in[26]; const float* time_b = (const float*)d_in[27];
  const float* kW = (const float*)d_in[28]; const float* kB = (const float*)d_in[29];
  const float* qW = (const float*)d_in[30]; const float* qB = (const float*)d_in[31];
  const float* vW = (const float*)d_in[32]; const float* vB = (const float*)d_in[33];
  const float* aW = (const float*)d_in[34]; const float* aB = (const float*)d_in[35];
  const float* skip = (const float*)d_in[36];
  const float* ln_g = (const float*)d_in[37]; const float* ln_b = (const float*)d_in[38];
  const float* pri  = (const float*)d_in[39];
  const float* attR = (const float*)d_in[40]; const float* msgR = (const float*)d_in[41];
  const float* out_w= (const float*)d_in[42]; const float* out_b= (const float*)d_in[43];

  const int NW = in_sizes[0], NT = in_sizes[1], ND = in_sizes[4];
  const int Gn = in_sizes[19];
  const int Eww = in_sizes[5], Ewd = in_sizes[8], Ewt = in_sizes[11];
  const int Etd = in_sizes[14], Ett = in_sizes[17];
  int nN[3] = {NW, NT, ND};
  int maxN = NW; if (NT > maxN) maxN = NT; if (ND > maxN) maxN = ND;
  int maxE = Eww; if (Ewd > maxE) maxE = Ewd; if (Ewt > maxE) maxE = Ewt;
  if (Etd > maxE) maxE = Etd; if (Ett > maxE) maxE = Ett;

  // carve workspace
  float* w = (float*)d_ws;
  auto take = [&](size_t cnt) { float* p = w; w += cnt; return p; };
  float *Hb[3], *NHb[3], *Kb[3], *Qb[3], *Vb[3], *Tb[3];
  for (int i = 0; i < 3; ++i) Hb[i]  = take((size_t)nN[i] * NHID);
  for (int i = 0; i < 3; ++i) NHb[i] = take((size_t)nN[i] * NHID);
  for (int i = 0; i < 3; ++i) Kb[i]  = take((size_t)nN[i] * NHID);
  for (int i = 0; i < 3; ++i) Qb[i]  = take((size_t)nN[i] * NHID);
  for (int i = 0; i < 3; ++i) Vb[i]  = take((size_t)nN[i] * NHID);
  for (int i = 0; i < 3; ++i) Tb[i]  = take((size_t)nN[i] * NHID);
  float* QA    = take((size_t)maxN * NHID);   // also reused as 'trans' buffer
  float* VM    = take((size_t)maxN * NHID);
  float* AGG   = take((size_t)maxN * NHID);
  float* AEDGE = take((size_t)maxE * HEADS);
  float* DEN   = take((size_t)maxN * HEADS);
  float* THT   = take(7 * 32);
  unsigned int* MBUF = (unsigned int*)take((size_t)maxN * HEADS);
  unsigned int* POOL = (unsigned int*)take((size_t)Gn * 3 * NHID);

  const dim3 blk(256);

  // temporal encoding table + initial node features
  time_enc_k<<<1, 256, 0, stream>>>(time_tab, time_w, time_b, THT);
  gemm256_wmma_k<<<dim3(4, cdiv(NW, 128)), blk, 0, stream>>>(
      word_embeds, word_id, adapt_w, adapt_b, Hb[0], NW, 1.0f);
  gather_rows_k<<<cdiv((long)NT * NHID, 256), blk, 0, stream>>>(topic_embeds, topic_id, Hb[1], NT);
  gather_rows_k<<<cdiv((long)ND * NHID, 256), blk, 0, stream>>>(doc_gen, nullptr, Hb[2], ND);

  struct RelT { int e, s, d; const int *src, *dst, *time; int E; };
  RelT rels[8] = {
      {0, 0, 1, src_wt, dst_wt, time_wt, Ewt},
      {1, 0, 2, src_wd, dst_wd, time_wd, Ewd},
      {2, 1, 2, src_td, dst_td, time_td, Etd},
      {3, 1, 1, src_tt, dst_tt, nullptr, Ett},
      {4, 0, 0, src_ww, dst_ww, time_ww, Eww},
      {5, 1, 0, dst_wt, src_wt, time_wt, Ewt},
      {6, 2, 1, dst_td, src_td, time_td, Etd},
      {7, 2, 0, dst_wd, src_wd, time_wd, Ewd},
  };
  const float ninv[3] = {1.0f / 3.0f, 1.0f / 3.0f, 1.0f / 2.0f};

  for (int li = 0; li < 2; ++li) {
    for (int i = 0; i < 3; ++i) {
      const size_t wo = (size_t)(li * 3 + i) * NHID * NHID;
      const size_t bo = (size_t)(li * 3 + i) * NHID;
      const dim3 grd(4, cdiv(nN[i], 128));
      gemm256_wmma_k<<<grd, blk, 0, stream>>>(Hb[i], nullptr, kW + wo, kB + bo, Kb[i], nN[i], 1.0f);
      gemm256_wmma_k<<<grd, blk, 0, stream>>>(Hb[i], nullptr, qW + wo, qB + bo, Qb[i], nN[i], 1.0f);
      gemm256_wmma_k<<<grd, blk, 0, stream>>>(Hb[i], nullptr, vW + wo, vB + bo, Vb[i], nN[i], 1.0f);
      fill_f32_k<<<cdiv((long)nN[i] * NHID, 256), blk, 0, stream>>>(Tb[i], 0.0f, (long)nN[i] * NHID);
    }
    for (int r = 0; r < 8; ++r) {
      const RelT& R = rels[r];
      const size_t ro = (size_t)(li * 8 + R.e) * HEADS * DKv * DKv;
      headgemm_wmma_k<<<dim3(cdiv(nN[R.d], 128), HEADS), blk, 0, stream>>>(Qb[R.d], attR + ro, QA, nN[R.d], 1);
      headgemm_wmma_k<<<dim3(cdiv(nN[R.s], 128), HEADS), blk, 0, stream>>>(Vb[R.s], msgR + ro, VM, nN[R.s], 0);
      fill_u32_k<<<cdiv((long)nN[R.d] * HEADS, 256), blk, 0, stream>>>(MBUF, ORD_NEG_INF, (long)nN[R.d] * HEADS);
      fill_f32_k<<<cdiv((long)nN[R.d] * HEADS, 256), blk, 0, stream>>>(DEN, 0.0f, (long)nN[R.d] * HEADS);
      fill_f32_k<<<cdiv((long)nN[R.d] * NHID, 256), blk, 0, stream>>>(AGG, 0.0f, (long)nN[R.d] * NHID);
      edge_score_k<<<cdiv((long)R.E * HEADS, 256), blk, 0, stream>>>(
          Kb[R.s], QA, R.src, R.dst, pri + (size_t)(li * 8 + R.e) * HEADS, AEDGE, MBUF, R.E);
      edge_exp_k<<<cdiv((long)R.E * HEADS, 256), blk, 0, stream>>>(R.dst, MBUF, AEDGE, DEN, R.E);
      edge_agg_k<<<cdiv((long)R.E * NHID, 256), blk, 0, stream>>>(
          R.src, R.dst, R.time, THT, VM, AEDGE, DEN, AGG, R.E);
      relu_acc_k<<<cdiv((long)nN[R.d] * NHID, 256), blk, 0, stream>>>(AGG, Tb[R.d], (long)nN[R.d] * NHID);
    }
    for (int i = 0; i < 3; ++i) {
      const size_t wo = (size_t)(li * 3 + i) * NHID * NHID;
      const size_t bo = (size_t)(li * 3 + i) * NHID;
      gemm256_wmma_k<<<dim3(4, cdiv(nN[i], 128)), blk, 0, stream>>>(
          Tb[i], nullptr, aW + wo, aB + bo, QA, nN[i], ninv[i]);
      skip_ln_k<<<nN[i], blk, 0, stream>>>(QA, Hb[i], skip + li * 3 + i, ln_g + bo, ln_b + bo, NHb[i]);
    }
    for (int i = 0; i < 3; ++i) { float* tmp = Hb[i]; Hb[i] = NHb[i]; NHb[i] = tmp; }
  }

  // graph max-pool: gi = [doc | word | topic]
  fill_u32_k<<<cdiv((long)Gn * 3 * NHID, 256), blk, 0, stream>>>(POOL, ORD_NEG_INF, (long)Gn * 3 * NHID);
  pool_max_k<<<cdiv((long)ND * NHID, 256), blk, 0, stream>>>(Hb[2], g_doc,   POOL, ND, 0);
  pool_max_k<<<cdiv((long)NW * NHID, 256), blk, 0, stream>>>(Hb[0], g_word,  POOL, NW, NHID);
  pool_max_k<<<cdiv((long)NT * NHID, 256), blk, 0, stream>>>(Hb[1], g_topic, POOL, NT, 2 * NHID);
  final_k<<<1, 64, 0, stream>>>(POOL, out_w, out_b, y_data, (float*)d_out, Gn);
}